// EnhancedFRAUnifiedEncoder_18476949307922
// MI455X (gfx1250) — compile-verified
//
#include <hip/hip_runtime.h>
#include <hip/hip_bf16.h>

#define BATCH 8
#define NNODE 2048
#define HID   512
#define KNN   8
#define MAXD  64
#define NROWS (BATCH * NNODE)   // 16384

typedef float v2f __attribute__((ext_vector_type(2)));
typedef float v8f __attribute__((ext_vector_type(8)));

// ---------------------------------------------------------------------------
// 1) k-NN: one thread per node, coords of the batch staged in LDS.
//    grid = 64 blocks (8 per batch), block = 256 threads.
// ---------------------------------------------------------------------------
__global__ __launch_bounds__(256) void knn_kernel(const float* __restrict__ coords,
                                                  int* __restrict__ knn,
                                                  int* __restrict__ cnt,
                                                  int* __restrict__ nbr) {
    __shared__ float2 sc[NNODE];
    const int b  = blockIdx.x >> 3;
    const int i0 = (blockIdx.x & 7) * 256;
    const float2* cb = (const float2*)(coords + (size_t)b * NNODE * 2);
    for (int j = threadIdx.x; j < NNODE; j += 256) sc[j] = cb[j];
    __syncthreads();

    const int i = i0 + threadIdx.x;
    const float2 ci = sc[i];

    float bd[KNN]; int bi[KNN];
#pragma unroll
    for (int k = 0; k < KNN; ++k) { bd[k] = 3.4e38f; bi[k] = 0; }

    for (int j = 0; j < NNODE; ++j) {
        if (j == i) continue;
        const float dx = sc[j].x - ci.x;
        const float dy = sc[j].y - ci.y;
        const float d2 = dx * dx + dy * dy;
        if (d2 < bd[KNN - 1]) {
            float dc = d2; int ic = j;
#pragma unroll
            for (int k = 0; k < KNN; ++k) {
                if (dc < bd[k]) {
                    const float td = bd[k]; const int ti = bi[k];
                    bd[k] = dc; bi[k] = ic; dc = td; ic = ti;
                }
            }
        }
    }

    const int row = b * NNODE + i;
    cnt[row] = KNN;   // re-initialized every call => deterministic
#pragma unroll
    for (int k = 0; k < KNN; ++k) {
        knn[(size_t)row * KNN + k]  = bi[k];
        nbr[(size_t)row * MAXD + k] = bi[k];
    }
}

// ---------------------------------------------------------------------------
// 2) Symmetrize: append reverse edges (dedup against target's own knn list).
//    One thread per directed edge (B*N*K = 131072).
// ---------------------------------------------------------------------------
__global__ __launch_bounds__(256) void reverse_kernel(const int* __restrict__ knn,
                                                      int* __restrict__ cnt,
                                                      int* __restrict__ nbr) {
    const int gid = blockIdx.x * 256 + threadIdx.x;
    if (gid >= NROWS * KNN) return;
    const int row = gid >> 3;          // source node (b, j)
    const int b   = row >> 11;
    const int j   = row & (NNODE - 1);
    const int t   = knn[gid];          // target node (within batch)
    const int trow = (b << 11) + t;

    bool dup = false;
#pragma unroll
    for (int k = 0; k < KNN; ++k) dup |= (knn[(size_t)trow * KNN + k] == j);
    if (!dup) {
        const int pos = atomicAdd(&cnt[trow], 1);
        if (pos < MAXD) nbr[(size_t)trow * MAXD + pos] = j;
    }
}

// ---------------------------------------------------------------------------
// 3) Sparse aggregation: y[row] = sum over neighbors of x[nbr].
//    One wave32 per row; lane owns 16 floats as 4x float4.
//    grid = 2048 blocks of 256 (8 waves).
// ---------------------------------------------------------------------------
__global__ __launch_bounds__(256) void aggregate_kernel(const float* __restrict__ x,
                                                        const int* __restrict__ nbr,
                                                        const int* __restrict__ cnt,
                                                        float* __restrict__ y) {
    const int wave = threadIdx.x >> 5;
    const int lane = threadIdx.x & 31;
    const int row  = blockIdx.x * 8 + wave;
    const int b    = row >> 11;
    const float* xb = x + (size_t)b * NNODE * HID;
    const int* lst  = nbr + (size_t)row * MAXD;
    int d = cnt[row];
    if (d > MAXD) d = MAXD;

    float4 acc[4] = {};
    for (int e = 0; e < d; ++e) {
        const float* xr = xb + (size_t)lst[e] * HID;
        if (e + 1 < d) __builtin_prefetch(xb + (size_t)lst[e + 1] * HID + lane * 4, 0, 0);
#pragma unroll
        for (int q = 0; q < 4; ++q) {
            const float4 v = *(const float4*)(xr + q * 128 + lane * 4);
            acc[q].x += v.x; acc[q].y += v.y; acc[q].z += v.z; acc[q].w += v.w;
        }
    }
    float* yr = y + (size_t)row * HID;
#pragma unroll
    for (int q = 0; q < 4; ++q) *(float4*)(yr + q * 128 + lane * 4) = acc[q];
}

// ---------------------------------------------------------------------------
// 4) GEMM  Z = Y @ W  via V_WMMA_F32_16X16X4_F32 (exact fp32 matrix path).
//    Block = 256 threads = 8 waves. Each wave owns a 32(M) x 64(N) tile
//    (2 M-fragments x 4 N-fragments = 8 accumulators) so every B-fragment
//    fetched from LDS feeds TWO wmmas: 0.5 ds-loads + 0.25 global-loads
//    per v_wmma. Block tile 256(M) x 64(N), K staged in 64-wide chunks.
//    W chunk staged TRANSPOSED in LDS (Bs[n][k], stride 68 -> conflict-free
//    float2 DS reads). grid = (512/64, 16384/256) = (8, 64).
// ---------------------------------------------------------------------------
__global__ __launch_bounds__(256) void gemm_wmma(const float* __restrict__ A,
                                                 const float* __restrict__ Bw,
                                                 float* __restrict__ C) {
    __shared__ float Bs[64][68];

    const int wave  = threadIdx.x >> 5;
    const int lane  = threadIdx.x & 31;
    const int mlane = lane & 15;
    const int h     = lane >> 4;
    const int m0    = blockIdx.y * 256 + wave * 32;
    const int n0    = blockIdx.x * 64;

    v8f acc[2][4] = {};
    const float* Arow0 = A + (size_t)(m0 + mlane) * HID;
    const float* Arow1 = A + (size_t)(m0 + 16 + mlane) * HID;

    for (int k0 = 0; k0 < HID; k0 += 64) {
        // cooperative transposed load of W[k0:k0+64][n0:n0+64] into Bs[n][k]
        {
            const int col4 = (threadIdx.x & 15) * 4;
            const int kr   = threadIdx.x >> 4;   // 0..15
#pragma unroll
            for (int it = 0; it < 4; ++it) {
                const int k = kr + it * 16;
                const float4 v = *(const float4*)(Bw + (size_t)(k0 + k) * HID + n0 + col4);
                Bs[col4 + 0][k] = v.x;
                Bs[col4 + 1][k] = v.y;
                Bs[col4 + 2][k] = v.z;
                Bs[col4 + 3][k] = v.w;
            }
        }
        __syncthreads();

#pragma unroll
        for (int kk = 0; kk < 64; kk += 4) {
            // A fragments: lanes 0-15 hold K=kk,kk+1 ; lanes 16-31 hold K=kk+2,kk+3
            const float2 av0 = *(const float2*)(Arow0 + k0 + kk + 2 * h);
            const float2 av1 = *(const float2*)(Arow1 + k0 + kk + 2 * h);
            v2f a0; a0.x = av0.x; a0.y = av0.y;
            v2f a1; a1.x = av1.x; a1.y = av1.y;
#pragma unroll
            for (int c = 0; c < 4; ++c) {
                const float2 bv = *(const float2*)(&Bs[c * 16 + mlane][kk + 2 * h]);
                v2f bf; bf.x = bv.x; bf.y = bv.y;
                acc[0][c] = __builtin_amdgcn_wmma_f32_16x16x4_f32(
                    false, a0, false, bf, (short)0, acc[0][c], false, false);
                acc[1][c] = __builtin_amdgcn_wmma_f32_16x16x4_f32(
                    false, a1, false, bf, (short)0, acc[1][c], false, false);
            }
        }
        __syncthreads();
    }

    // D layout: VGPR r -> lanes 0-15: M=r, N=lane ; lanes 16-31: M=8+r, N=lane-16
#pragma unroll
    for (int t = 0; t < 2; ++t) {
#pragma unroll
        for (int c = 0; c < 4; ++c) {
#pragma unroll
            for (int r = 0; r < 8; ++r) {
                const int m = m0 + t * 16 + r + 8 * h;
                const int n = n0 + c * 16 + mlane;
                C[(size_t)m * HID + n] = acc[t][c][r];
            }
        }
    }
}

// ---------------------------------------------------------------------------
// 5) Epilogue: bias + LayerNorm + ReLU (+ residual), one wave32 per row.
//    grid = 2048 blocks of 256.
// ---------------------------------------------------------------------------
__global__ __launch_bounds__(256) void ln_relu_res_kernel(const float* __restrict__ Z,
                                                          const float* __restrict__ bias,
                                                          const float* __restrict__ gamma,
                                                          const float* __restrict__ beta,
                                                          float* __restrict__ x,
                                                          int residual) {
    const int wave = threadIdx.x >> 5;
    const int lane = threadIdx.x & 31;
    const int row  = blockIdx.x * 8 + wave;
    const float* zr = Z + (size_t)row * HID;

    float v[16];
    float s = 0.f, s2 = 0.f;
#pragma unroll
    for (int q = 0; q < 4; ++q) {
        const int col = q * 128 + lane * 4;
        const float4 t  = *(const float4*)(zr + col);
        const float4 bb = *(const float4*)(bias + col);
        v[q * 4 + 0] = t.x + bb.x;
        v[q * 4 + 1] = t.y + bb.y;
        v[q * 4 + 2] = t.z + bb.z;
        v[q * 4 + 3] = t.w + bb.w;
#pragma unroll
        for (int e = 0; e < 4; ++e) { const float f = v[q * 4 + e]; s += f; s2 += f * f; }
    }
#pragma unroll
    for (int off = 16; off > 0; off >>= 1) {
        s  += __shfl_xor(s, off, 32);
        s2 += __shfl_xor(s2, off, 32);
    }
    const float mu  = s * (1.f / HID);
    const float var = s2 * (1.f / HID) - mu * mu;
    const float rs  = rsqrtf(var + 1e-5f);

    float* xr = x + (size_t)row * HID;
#pragma unroll
    for (int q = 0; q < 4; ++q) {
        const int col = q * 128 + lane * 4;
        const float4 g  = *(const float4*)(gamma + col);
        const float4 bt = *(const float4*)(beta + col);
        float4 prev = {0.f, 0.f, 0.f, 0.f};
        if (residual) prev = *(const float4*)(xr + col);
        float4 o;
        o.x = fmaxf((v[q * 4 + 0] - mu) * rs * g.x + bt.x, 0.f) + prev.x;
        o.y = fmaxf((v[q * 4 + 1] - mu) * rs * g.y + bt.y, 0.f) + prev.y;
        o.z = fmaxf((v[q * 4 + 2] - mu) * rs * g.z + bt.z, 0.f) + prev.z;
        o.w = fmaxf((v[q * 4 + 3] - mu) * rs * g.w + bt.w, 0.f) + prev.w;
        *(float4*)(xr + col) = o;
    }
}

// ---------------------------------------------------------------------------
extern "C" void kernel_launch(void* const* d_in, const int* in_sizes, int n_in,
                              void* d_out, int out_size, void* d_ws, size_t ws_size,
                              hipStream_t stream) {
    const float* node   = (const float*)d_in[0];   // [8,2048,512]
    const float* coords = (const float*)d_in[1];   // [8,2048,2]
    const float* W      = (const float*)d_in[2];   // [3,512,512]
    const float* bias   = (const float*)d_in[3];   // [3,512]
    const float* gamma  = (const float*)d_in[4];   // [3,512]
    const float* beta   = (const float*)d_in[5];   // [3,512]
    float* out = (float*)d_out;                    // doubles as persistent x

    char* ws = (char*)d_ws;
    size_t off = 0;
    int* knn = (int*)(ws + off); off += (size_t)NROWS * KNN * sizeof(int);
    int* cnt = (int*)(ws + off); off += (size_t)NROWS * sizeof(int);
    int* nbr = (int*)(ws + off); off += (size_t)NROWS * MAXD * sizeof(int);
    off = (off + 255) & ~(size_t)255;
    float* yagg = (float*)(ws + off); off += (size_t)NROWS * HID * sizeof(float);
    float* z    = (float*)(ws + off);

    knn_kernel<<<64, 256, 0, stream>>>(coords, knn, cnt, nbr);
    reverse_kernel<<<(NROWS * KNN + 255) / 256, 256, 0, stream>>>(knn, cnt, nbr);

    for (int l = 0; l < 3; ++l) {
        const float* xin = (l == 0) ? node : out;
        aggregate_kernel<<<NROWS / 8, 256, 0, stream>>>(xin, nbr, cnt, yagg);
        gemm_wmma<<<dim3(HID / 64, NROWS / 256), 256, 0, stream>>>(yagg, W + (size_t)l * HID * HID, z);
        ln_relu_res_kernel<<<NROWS / 8, 256, 0, stream>>>(
            z, bias + l * HID, gamma + l * HID, beta + l * HID, out, l > 0 ? 1 : 0);
    }
}